// Rope_63685775065637
// MI455X (gfx1250) — compile-verified
//
#include <hip/hip_runtime.h>
#include <hip/hip_bf16.h>

// RoPE: out = rotate_pairs(x, angle(b,s,k)), x:(B,H,S,D) f32, pos:(B,S) i64.
// Memory-bound (AI ~ 1 flop/byte): 134 MB of traffic -> ~5.8us floor at
// 23.3 TB/s. Strategy: b128 coalesced streaming, sincos computed once per
// (s,k) and shared across the 16 heads via LDS, loads issued ahead of the
// transcendental phase so HBM latency overlaps the sin/cos reduction.

namespace {
constexpr int kB  = 4;
constexpr int kH  = 16;
constexpr int kS  = 4096;
constexpr int kD  = 64;      // head dim (floats per row)
constexpr int kKP = kD / 2;  // 32 rotation pairs per row
constexpr int kNS = 4;       // s-positions handled per block
constexpr int kV4 = kD / 4;  // float4 per row = 16
}

__global__ __launch_bounds__(256)
void rope_kernel(const float* __restrict__ x,
                 const int* __restrict__ pos_lo,   // i64 positions, viewed as i32 pairs
                 float* __restrict__ out)
{
    __shared__ float s_cos[kNS][kKP];
    __shared__ float s_sin[kNS][kKP];

    const int tid = threadIdx.x;
    const int bid = blockIdx.x;
    const int blocks_per_b = kS / kNS;            // 1024
    const int b  = bid / blocks_per_b;
    const int s0 = (bid % blocks_per_b) * kNS;

    // Thread -> (head, float4-slot-within-row). Each float4 = 2 rotation pairs.
    const int h  = tid >> 4;                      // 0..15
    const int d4 = tid & 15;                      // 0..15
    const int k0 = d4 * 2;

    const float4* __restrict__ x4 = (const float4*)x;
    float4*       __restrict__ o4 = (float4*)out;

    // float4 index of (b,h,s0) row start, plus this thread's slot in the row.
    const long long rowbase =
        ((long long)(b * kH + h) * kS + s0) * kV4 + d4;

    // ---- Issue all tile loads up front: 4 outstanding global_load_b128 per
    // thread (16 KB/block in flight) overlap with the sincos phase below;
    // s_wait_loadcnt lands just before first use.
    float4 v[kNS];
    #pragma unroll
    for (int si = 0; si < kNS; ++si)
        v[si] = x4[rowbase + (long long)si * kV4];

    // ---- Compute sin/cos once per (si,k); reused by all 16 heads (16x less
    // transcendental VALU). Positions < 8192, so the low dword of the
    // little-endian i64 is exact -> cheap v_cvt_f32_i32 instead of a 64-bit cvt.
    if (tid < kNS * kKP) {
        const int si = tid >> 5;                  // 0..3
        const int k  = tid & 31;                  // 0..31
        const float p = (float)pos_lo[2 * (b * kS + (s0 + si))];
        // rate_k = 10000^(-2k/64) = exp2(k * -log2(10000)/32)
        const float C = -0.41524101186093026f;    // -log2(10000)/32
        const float angle = p * exp2f((float)k * C);
        float sv, cv;
        sincosf(angle, &sv, &cv);
        s_cos[si][k] = cv;
        s_sin[si][k] = sv;
    }
    __syncthreads();

    // ---- Rotate and stream out.
    #pragma unroll
    for (int si = 0; si < kNS; ++si) {
        const float c0  = s_cos[si][k0];
        const float sn0 = s_sin[si][k0];
        const float c1  = s_cos[si][k0 + 1];
        const float sn1 = s_sin[si][k0 + 1];
        float4 r;
        r.x = c0  * v[si].x - sn0 * v[si].y;
        r.y = sn0 * v[si].x + c0  * v[si].y;
        r.z = c1  * v[si].z - sn1 * v[si].w;
        r.w = sn1 * v[si].z + c1  * v[si].w;
        o4[rowbase + (long long)si * kV4] = r;    // global_store_b128
    }
}

extern "C" void kernel_launch(void* const* d_in, const int* in_sizes, int n_in,
                              void* d_out, int out_size, void* d_ws, size_t ws_size,
                              hipStream_t stream) {
    const float* x      = (const float*)d_in[0];
    const int*   pos_lo = (const int*)d_in[1];    // int64 positions, low dwords
    float*       out    = (float*)d_out;

    const int grid = (kB * kS) / kNS;   // 4096 blocks of 256 threads (8 waves)
    rope_kernel<<<grid, 256, 0, stream>>>(x, pos_lo, out);
}